// GIKT_PYG_15152644620331
// MI455X (gfx1250) — compile-verified
//
#include <hip/hip_runtime.h>

// ---------------- problem constants ----------------
#define Bsz 64
#define Ssz 200
#define Dsz 128
#define NQc 50000
#define NCc 1000
#define MCc 4
#define Kk  10

typedef float v2f __attribute__((ext_vector_type(2)));
typedef float v8f __attribute__((ext_vector_type(8)));

// ---------------------------------------------------------------------------
// Wave-level f32 WMMA helper: computes one 16x16 tile of  As[16xK] @ W^T
// where W is row-major [Nrows x ldw] and we take rows [colbase, colbase+16).
// A-tile layout (ISA 7.12.2, 32-bit A 16x4): lane l: m=l&15, k=2*(l>>4)+{0,1}
// B-tile (K x 16, B[k][n] = W[colbase+n][k]) mirrors the same per-lane pattern.
// ---------------------------------------------------------------------------
__device__ __forceinline__ v8f wmma_k(const float* __restrict__ As, int lda,
                                      const float* __restrict__ W, int ldw,
                                      int colbase, int K, int lane) {
  const int m   = lane & 15;
  const int khi = (lane >> 4) << 1;
  const float* ar = As + m * lda + khi;
  const float* wr = W + (size_t)(colbase + (lane & 15)) * ldw + khi;
  v8f c = {};
  for (int k0 = 0; k0 < K; k0 += 4) {
    v2f a; a.x = ar[k0]; a.y = ar[k0 + 1];
    v2f b; b.x = wr[k0]; b.y = wr[k0 + 1];
    c = __builtin_amdgcn_wmma_f32_16x16x4_f32(false, a, false, b,
                                              (short)0, c, false, false);
  }
  return c;
}

// C/D layout: VGPR r -> row r + 8*(lane>>4), col = lane&15
__device__ __forceinline__ void store_bias(float* O, int ldo, int coloff,
                                           const float* bias, v8f c, int lane) {
  const int col   = coloff + (lane & 15);
  const int rbase = (lane >> 4) << 3;
  const float bv  = bias ? bias[col] : 0.f;
#pragma unroll
  for (int r = 0; r < 8; ++r) O[(rbase + r) * ldo + col] = c[r] + bv;
}

// ---------------------------------------------------------------------------
// Generic  out[Mx128] = act( A[Mx128] @ W[128x128]^T + bias )
// 256 threads = 8 waves; block = 16 rows, wave w = column tile w.
// ---------------------------------------------------------------------------
__global__ void __launch_bounds__(256)
gemm128_kernel(const float* __restrict__ A, const float* __restrict__ W,
               const float* __restrict__ bias, float* __restrict__ out,
               int M, int act) {
  __shared__ float As[16 * 128];
  const int tid = threadIdx.x, lane = tid & 31, wave = tid >> 5;
  const int row0 = blockIdx.x << 4;
  for (int i = tid; i < 16 * 128; i += 256) {
    int r = i >> 7, cc = i & 127;
    int gr = row0 + r;
    As[i] = (gr < M) ? A[(size_t)gr * 128 + cc] : 0.f;
  }
  __syncthreads();
  v8f c = wmma_k(As, 128, W, 128, wave * 16, 128, lane);
  const int col   = wave * 16 + (lane & 15);
  const int rbase = (lane >> 4) << 3;
  const float bv  = bias ? bias[col] : 0.f;
#pragma unroll
  for (int r = 0; r < 8; ++r) {
    int rowg = row0 + rbase + r;
    if (rowg < M) {
      float v = c[r] + bv;
      if (act) v = tanhf(v);
      out[(size_t)rowg * 128 + col] = v;
    }
  }
}

// ---------------- GNN helper kernels ----------------
__global__ void zero_agg(long long* sumC, int* cnt) {
  int i = blockIdx.x * 256 + threadIdx.x;
  if (i < NCc * 128) sumC[i] = 0;
  if (i < NCc) cnt[i] = 0;
}

__global__ void count_kernel(const int* __restrict__ qtab, int* cnt) {
  int e = blockIdx.x * 256 + threadIdx.x;
  if (e < NQc * MCc) atomicAdd(&cnt[qtab[e]], 1);
}

// preQ1[q] = emb_q[q] + mean_c emb_c   (each question has exactly MC=4 edges)
__global__ void preq1_kernel(const float* __restrict__ emb_q,
                             const float* __restrict__ emb_c,
                             const int* __restrict__ qtab, float* outp) {
  int q = blockIdx.x, d = threadIdx.x;
  const int* qt = qtab + q * 4;
  float s = emb_c[(size_t)qt[0] * 128 + d] + emb_c[(size_t)qt[1] * 128 + d] +
            emb_c[(size_t)qt[2] * 128 + d] + emb_c[(size_t)qt[3] * 128 + d];
  outp[(size_t)q * 128 + d] = emb_q[(size_t)q * 128 + d] + 0.25f * s;
}

// deterministic scatter: 2^36 fixed point, u64 integer atomics (order independent)
#define FXSCALE 68719476736.0f
__global__ void scatter_kernel(const float* __restrict__ x1q,
                               const int* __restrict__ qtab, long long* sumC) {
  int q = blockIdx.x, d = threadIdx.x;
  float v = x1q[(size_t)q * 128 + d];
  long long fx = (long long)rintf(v * FXSCALE);
#pragma unroll
  for (int m = 0; m < 4; ++m) {
    int c2 = qtab[q * 4 + m];
    atomicAdd((unsigned long long*)&sumC[(size_t)c2 * 128 + d],
              (unsigned long long)fx);
  }
}

__global__ void prec2_kernel(const float* __restrict__ x1c,
                             const long long* __restrict__ sumC,
                             const int* __restrict__ cnt, float* outp) {
  int c = blockIdx.x, d = threadIdx.x;
  float s = (float)((double)sumC[(size_t)c * 128 + d] * (1.0 / (double)FXSCALE));
  int cn = cnt[c]; if (cn < 1) cn = 1;
  outp[(size_t)c * 128 + d] = x1c[(size_t)c * 128 + d] + s / (float)cn;
}

__global__ void preq3_kernel(const float* __restrict__ x2q,
                             const float* __restrict__ x2c,
                             const int* __restrict__ qtab, float* outp) {
  int q = blockIdx.x, d = threadIdx.x;
  const int* qt = qtab + q * 4;
  float s = x2c[(size_t)qt[0] * 128 + d] + x2c[(size_t)qt[1] * 128 + d] +
            x2c[(size_t)qt[2] * 128 + d] + x2c[(size_t)qt[3] * 128 + d];
  outp[(size_t)q * 128 + d] = x2q[(size_t)q * 128 + d] + 0.25f * s;
}

__device__ __forceinline__ float sigmoidf(float x) {
  return 1.f / (1.f + __expf(-x));
}

// ---------------------------------------------------------------------------
// Sequential scan kernel: 4 blocks x 256 threads, 16 samples per block.
// Whole 199-step GRU+attention scan per block; WMMA f32 for all projections.
// ---------------------------------------------------------------------------
__global__ void __launch_bounds__(256)
seq_kernel(const float* __restrict__ xg, const float* __restrict__ emb_q,
           const float* __restrict__ emb_c, const float* __restrict__ emb_r,
           const float* __restrict__ Wih1, const float* __restrict__ Whh1,
           const float* __restrict__ bih1, const float* __restrict__ bhh1,
           const float* __restrict__ Wih2, const float* __restrict__ Whh2,
           const float* __restrict__ bih2, const float* __restrict__ bhh2,
           const float* __restrict__ Wq, const float* __restrict__ bq,
           const float* __restrict__ Wk, const float* __restrict__ bk,
           const float* __restrict__ Ww, const float* __restrict__ bw,
           const int* __restrict__ qseq, const int* __restrict__ rseq,
           const int* __restrict__ mseq, const int* __restrict__ qtab,
           float* __restrict__ keyhist, float* __restrict__ statehist,
           float* __restrict__ scorebuf, float* __restrict__ out) {
  extern __shared__ float sm[];
  float* xcat = sm;                    // 16*256
  float* h1   = xcat + 16 * 256;       // 16*128
  float* h2   = h1 + 16 * 128;         // 16*128
  float* gi   = h2 + 16 * 128;         // 16*384
  float* gh   = gi + 16 * 384;         // 16*384
  float* qc   = gh + 16 * 384;         // 80*128
  float* aq   = qc + 80 * 128;         // 80*128
  float* topv = aq + 80 * 128;         // 16*10
  int*   topi = (int*)(topv + 160);    // 16*10

  const int tid = threadIdx.x, lane = tid & 31, wave = tid >> 5;
  const int b0 = blockIdx.x * 16;
  float* myscore = scorebuf + (size_t)blockIdx.x * (16 * 5 * Ssz);

  for (int i = tid; i < 16 * 128; i += 256) { h1[i] = 0.f; h2[i] = 0.f; }
  if (tid < 16) out[(size_t)(b0 + tid) * Ssz] = 0.f;  // y_hat[:,0] = 0
  __syncthreads();

  for (int t = 0; t < Ssz - 1; ++t) {
    // ---- A: gather [e_q | e_r] ----
    for (int i = tid; i < 16 * 256; i += 256) {
      int s = i >> 8, d = i & 255;
      int b = b0 + s;
      float v;
      if (d < 128) {
        int q = qseq[b * Ssz + t];
        int m = mseq[b * Ssz + t];
        v = m ? xg[(size_t)q * 128 + d] : emb_q[(size_t)q * 128 + d];
      } else {
        int r = rseq[b * Ssz + t];
        v = emb_r[(size_t)r * 128 + (d - 128)];
      }
      xcat[i] = v;
    }
    __syncthreads();
    // ---- B: GRU1 gates (WMMA) ----
    for (int ct = wave; ct < 24; ct += 8) {
      v8f c  = wmma_k(xcat, 256, Wih1, 256, ct * 16, 256, lane);
      store_bias(gi, 384, ct * 16, bih1, c, lane);
      v8f c2 = wmma_k(h1, 128, Whh1, 128, ct * 16, 128, lane);
      store_bias(gh, 384, ct * 16, bhh1, c2, lane);
    }
    __syncthreads();
    // ---- C: gate math -> h1 ----
    for (int i = tid; i < 16 * 128; i += 256) {
      int s = i >> 7, d = i & 127;
      float r = sigmoidf(gi[s * 384 + d] + gh[s * 384 + d]);
      float z = sigmoidf(gi[s * 384 + 128 + d] + gh[s * 384 + 128 + d]);
      float n = tanhf(gi[s * 384 + 256 + d] + r * gh[s * 384 + 256 + d]);
      h1[i] = (1.f - z) * n + z * h1[i];
    }
    __syncthreads();
    // ---- D: GRU2 gates (WMMA) ----
    for (int ct = wave; ct < 24; ct += 8) {
      v8f c  = wmma_k(h1, 128, Wih2, 128, ct * 16, 128, lane);
      store_bias(gi, 384, ct * 16, bih2, c, lane);
      v8f c2 = wmma_k(h2, 128, Whh2, 128, ct * 16, 128, lane);
      store_bias(gh, 384, ct * 16, bhh2, c2, lane);
    }
    __syncthreads();
    // ---- E: gate math -> h2 + statehist ----
    for (int i = tid; i < 16 * 128; i += 256) {
      int s = i >> 7, d = i & 127;
      float r = sigmoidf(gi[s * 384 + d] + gh[s * 384 + d]);
      float z = sigmoidf(gi[s * 384 + 128 + d] + gh[s * 384 + 128 + d]);
      float n = tanhf(gi[s * 384 + 256 + d] + r * gh[s * 384 + 256 + d]);
      float v = (1.f - z) * n + z * h2[i];
      h2[i] = v;
      statehist[((size_t)(b0 + s) * Ssz + t) * 128 + d] = v;
    }
    __syncthreads();
    // ---- F: key_t = h2@Wk^T+bk -> keyhist  |  qc gather ----
    {
      v8f c = wmma_k(h2, 128, Wk, 128, wave * 16, 128, lane);
      int col = wave * 16 + (lane & 15);
      int rbase = (lane >> 4) << 3;
      float bv = bk[col];
#pragma unroll
      for (int r = 0; r < 8; ++r) {
        int s = rbase + r;
        keyhist[((size_t)(b0 + s) * Ssz + t) * 128 + col] = c[r] + bv;
      }
    }
    for (int i = tid; i < 80 * 128; i += 256) {
      int row = i >> 7, d = i & 127;
      int s = row / 5, ii = row - s * 5;
      int qn = qseq[(b0 + s) * Ssz + t + 1];
      qc[i] = (ii == 0) ? emb_q[(size_t)qn * 128 + d]
                        : emb_c[(size_t)qtab[qn * 4 + ii - 1] * 128 + d];
    }
    __syncthreads();
    // ---- G: aq = qc@Wq^T+bq (WMMA, 80 rows = 5 row tiles) ----
    for (int rt = 0; rt < 5; ++rt) {
      v8f c = wmma_k(qc + rt * 16 * 128, 128, Wq, 128, wave * 16, 128, lane);
      store_bias(aq + rt * 16 * 128, 128, wave * 16, bq, c, lane);
    }
    __syncthreads();
    // ---- H: scores[p][j] = aq[p] . keyhist[s][j], j<=t ----
    {
      int nval = t + 1;
      for (int p = wave; p < 80; p += 8) {
        int s = p / 5;
        const float4* av = (const float4*)(aq + p * 128);
        const float4* kb = (const float4*)(keyhist + (size_t)(b0 + s) * Ssz * 128);
        for (int j = lane; j < nval; j += 32) {
          const float4* kv = kb + (size_t)j * 32;
          float acc = 0.f;
#pragma unroll 8
          for (int d4 = 0; d4 < 32; ++d4) {
            float4 a = av[d4], k = kv[d4];
            acc += a.x * k.x + a.y * k.y + a.z * k.z + a.w * k.w;
          }
          myscore[p * Ssz + j] = acc;
        }
      }
    }
    __syncthreads();
    // ---- I: top-10 (16-lane argmax w/ register skip-list), softmax, readout ----
    {
      int nval = t + 1;
      int sub = lane >> 4, l16 = lane & 15;
      int s = wave * 2 + sub;
      const float* sbase = myscore + s * 5 * Ssz;
      int skipid[Kk];
      int nskip = 0;
#pragma unroll
      for (int u = 0; u < Kk; ++u) skipid[u] = -1;
      for (int k = 0; k < Kk; ++k) {
        float bv = -3.0e38f; int bi = -1;
        for (int i2 = 0; i2 < 5; ++i2) {
          int j0 = (l16 - i2 * nval) & 15;  // v = i2*nval + j  ==  l16 (mod 16)
          for (int j = j0; j < nval; j += 16) {
            int flat = i2 * Ssz + j;
            bool dup = false;
#pragma unroll
            for (int u = 0; u < Kk; ++u) dup |= (u < nskip) & (skipid[u] == flat);
            float v = sbase[flat];
            if (!dup && v > bv) { bv = v; bi = flat; }
          }
        }
        for (int off = 8; off >= 1; off >>= 1) {
          float ov = __shfl_xor(bv, off);
          int   oi = __shfl_xor(bi, off);
          if (ov > bv || (ov == bv && oi >= 0 && (bi < 0 || oi < bi))) { bv = ov; bi = oi; }
        }
        if (bi < 0) { bv = -1e9f; bi = 0; }  // pad (matches masked -1e9 entries)
        // register skip-list: owner lane of this candidate remembers it
        {
          int i2 = bi / Ssz, j = bi - i2 * Ssz;
          if (((i2 * nval + j) & 15) == l16 && nskip < Kk) {
#pragma unroll
            for (int u = 0; u < Kk; ++u) if (u == nskip) skipid[u] = bi;
            nskip++;
          }
        }
        if (l16 == 0) { topv[s * Kk + k] = bv; topi[s * Kk + k] = bi; }
      }
      // softmax over the 10 values + sigmoid readout (lane l16 = k)
      float myv = (l16 < Kk) ? topv[s * Kk + l16] : -3.0e38f;
      float mx = myv;
      for (int off = 8; off >= 1; off >>= 1) mx = fmaxf(mx, __shfl_xor(mx, off));
      float e = (l16 < Kk) ? __expf(myv - mx) : 0.f;
      float se = e;
      for (int off = 8; off >= 1; off >>= 1) se += __shfl_xor(se, off);
      float contrib = 0.f;
      if (l16 < Kk) {
        int fi = topi[s * Kk + l16];
        int i2 = fi / Ssz, j = fi - i2 * Ssz;
        const float* qrow = qc + (s * 5 + i2) * 128;
        const float* srow = statehist + ((size_t)(b0 + s) * Ssz + j) * 128;
        float acc = bw[0];
        for (int d = 0; d < 128; ++d) acc += qrow[d] * Ww[d];
        for (int d = 0; d < 128; ++d) acc += srow[d] * Ww[128 + d];
        contrib = (e / se) * sigmoidf(acc);
      }
      for (int off = 8; off >= 1; off >>= 1) contrib += __shfl_xor(contrib, off);
      if (l16 == 0) out[(size_t)(b0 + s) * Ssz + t + 1] = contrib;
    }
    __syncthreads();
  }
}

// ---------------------------------------------------------------------------
extern "C" void kernel_launch(void* const* d_in, const int* in_sizes, int n_in,
                              void* d_out, int out_size, void* d_ws, size_t ws_size,
                              hipStream_t stream) {
  const float* emb_q = (const float*)d_in[0];
  const float* emb_c = (const float*)d_in[1];
  const float* emb_r = (const float*)d_in[2];
  const float* Wih1  = (const float*)d_in[3];
  const float* Whh1  = (const float*)d_in[4];
  const float* bih1  = (const float*)d_in[5];
  const float* bhh1  = (const float*)d_in[6];
  const float* Wih2  = (const float*)d_in[7];
  const float* Whh2  = (const float*)d_in[8];
  const float* bih2  = (const float*)d_in[9];
  const float* bhh2  = (const float*)d_in[10];
  const float* Wg1   = (const float*)d_in[11];
  const float* Wg2   = (const float*)d_in[12];
  const float* Wg3   = (const float*)d_in[13];
  const float* Wagg  = (const float*)d_in[14];
  const float* bagg  = (const float*)d_in[15];
  const float* Wq    = (const float*)d_in[16];
  const float* bq    = (const float*)d_in[17];
  const float* Wk    = (const float*)d_in[18];
  const float* bk    = (const float*)d_in[19];
  const float* Ww    = (const float*)d_in[20];
  const float* bw    = (const float*)d_in[21];
  const int* qseq    = (const int*)d_in[22];
  const int* rseq    = (const int*)d_in[23];
  const int* mseq    = (const int*)d_in[24];
  const int* qtab    = (const int*)d_in[25];
  (void)in_sizes; (void)n_in; (void)out_size; (void)ws_size;

  char* ws = (char*)d_ws;
  size_t off = 0;
  auto carve = [&](size_t bytes) -> void* {
    void* p = ws + off;
    off += (bytes + 255) & ~(size_t)255;
    return p;
  };
  long long* sumC = (long long*)carve((size_t)NCc * 128 * 8);
  int* cnt        = (int*)carve((size_t)NCc * 4);
  float* buf0     = (float*)carve((size_t)NQc * 128 * 4);
  float* buf1     = (float*)carve((size_t)NQc * 128 * 4);
  float* xc0      = (float*)carve((size_t)NCc * 128 * 4);
  float* xc1      = (float*)carve((size_t)NCc * 128 * 4);
  float* keyh     = (float*)carve((size_t)Bsz * Ssz * 128 * 4);
  float* stah     = (float*)carve((size_t)Bsz * Ssz * 128 * 4);
  float* scoreb   = (float*)carve((size_t)4 * 16 * 5 * Ssz * 4);

  // ---- GNN: 3-hop SAGE + tanh(Wagg) on the 50k question rows ----
  zero_agg<<<(NCc * 128 + 255) / 256, 256, 0, stream>>>(sumC, cnt);
  count_kernel<<<(NQc * MCc + 255) / 256, 256, 0, stream>>>(qtab, cnt);
  preq1_kernel<<<NQc, 128, 0, stream>>>(emb_q, emb_c, qtab, buf0);
  gemm128_kernel<<<NQc / 16, 256, 0, stream>>>(buf0, Wg1, nullptr, buf1, NQc, 0);        // x1_q
  gemm128_kernel<<<(NCc + 15) / 16, 256, 0, stream>>>(emb_c, Wg1, nullptr, xc0, NCc, 0); // x1_c
  scatter_kernel<<<NQc, 128, 0, stream>>>(buf1, qtab, sumC);
  prec2_kernel<<<NCc, 128, 0, stream>>>(xc0, sumC, cnt, xc1);                            // preC2
  gemm128_kernel<<<(NCc + 15) / 16, 256, 0, stream>>>(xc1, Wg2, nullptr, xc0, NCc, 0);   // x2_c
  gemm128_kernel<<<NQc / 16, 256, 0, stream>>>(buf1, Wg2, nullptr, buf0, NQc, 0);        // x2_q
  preq3_kernel<<<NQc, 128, 0, stream>>>(buf0, xc0, qtab, buf1);                          // preQ3
  gemm128_kernel<<<NQc / 16, 256, 0, stream>>>(buf1, Wg3, nullptr, buf0, NQc, 0);        // x3_q
  gemm128_kernel<<<NQc / 16, 256, 0, stream>>>(buf0, Wagg, bagg, buf1, NQc, 1);          // xg=tanh

  // ---- sequential GRU + top-k attention scan ----
  size_t lds = (size_t)(16 * 256 + 16 * 128 * 2 + 16 * 384 * 2 + 80 * 128 * 2 + 320) * 4;
  seq_kernel<<<4, 256, lds, stream>>>(buf1, emb_q, emb_c, emb_r,
                                      Wih1, Whh1, bih1, bhh1,
                                      Wih2, Whh2, bih2, bhh2,
                                      Wq, bq, Wk, bk, Ww, bw,
                                      qseq, rseq, mseq, qtab,
                                      keyh, stah, scoreb, (float*)d_out);
}